// GIN_67800353734843
// MI455X (gfx1250) — compile-verified
//
#include <hip/hip_runtime.h>

typedef float v2f __attribute__((ext_vector_type(2)));
typedef float v8f __attribute__((ext_vector_type(8)));

#define N_NODES   100000
#define N_EDGES   1600000
#define N_GRAPHS  500
#define N_GRAPHS_PAD 512
#define DH        64
#define BN_EPS    1e-5f

__device__ __forceinline__ v8f wmma_f32_k4(v2f a, v2f b, v8f c) {
  // V_WMMA_F32_16X16X4_F32 : D = A(16x4 f32) x B(4x16 f32) + C(16x16 f32)
  return __builtin_amdgcn_wmma_f32_16x16x4_f32(
      /*neg_a=*/false, a, /*neg_b=*/false, b,
      /*c_mod=*/(short)0, c, /*reuse_a=*/false, /*reuse_b=*/false);
}

// ---------------- utility kernels ----------------

__global__ void vec_zero(float* __restrict__ p, int n) {
  int i = blockIdx.x * blockDim.x + threadIdx.x;
  if (i < n) p[i] = 0.0f;
}

__global__ void vec_copy4(float4* __restrict__ dst, const float4* __restrict__ src, int n4) {
  int i = blockIdx.x * blockDim.x + threadIdx.x;
  if (i < n4) dst[i] = src[i];
}

// agg[dst] += h[src] over all edges; 16 threads per edge, float4 each
__global__ void edge_scatter(const int* __restrict__ ei, const float* __restrict__ h,
                             float* __restrict__ agg, int nE) {
  int tid = blockIdx.x * blockDim.x + threadIdx.x;
  int e = tid >> 4;
  if (e >= nE) return;
  int p = (tid & 15) << 2;
  int s = ei[e];
  int d = ei[nE + e];
  float4 v = *(const float4*)(h + (size_t)s * DH + p);
  float* o = agg + (size_t)d * DH + p;
  atomicAdd(o + 0, v.x);
  atomicAdd(o + 1, v.y);
  atomicAdd(o + 2, v.z);
  atomicAdd(o + 3, v.w);
}

// pooled[batch[node]] += h[node]; 16 threads per node
__global__ void pool_scatter(const int* __restrict__ batch, const float* __restrict__ h,
                             float* __restrict__ pooled, int nN) {
  int tid = blockIdx.x * blockDim.x + threadIdx.x;
  int node = tid >> 4;
  if (node >= nN) return;
  int p = (tid & 15) << 2;
  int b = batch[node];
  float4 v = *(const float4*)(h + (size_t)node * DH + p);
  float* o = pooled + (size_t)b * DH + p;
  atomicAdd(o + 0, v.x);
  atomicAdd(o + 1, v.y);
  atomicAdd(o + 2, v.z);
  atomicAdd(o + 3, v.w);
}

// pad W2 (64x10) -> (64x16), b2 (10) -> (16) with zeros
__global__ void pad_w2(const float* __restrict__ W2, const float* __restrict__ b2,
                       float* __restrict__ W2p, float* __restrict__ b2p) {
  int i = blockIdx.x * blockDim.x + threadIdx.x;
  if (i < DH * 16) {
    int r = i >> 4, c = i & 15;
    W2p[i] = (c < 10) ? W2[r * 10 + c] : 0.0f;
  }
  if (i < 16) b2p[i] = (i < 10) ? b2[i] : 0.0f;
}

// finalize BN: fold (mu, var, gamma, beta) into per-feature scale/shift
__global__ void bn_finalize(const float* __restrict__ gsum, const float* __restrict__ gssq,
                            const float* __restrict__ g, const float* __restrict__ be,
                            float* __restrict__ scale, float* __restrict__ shift, float invN) {
  int i = threadIdx.x;
  float mu  = gsum[i] * invN;
  float var = gssq[i] * invN - mu * mu;
  float rs  = rsqrtf(var + BN_EPS);
  float sc  = rs * g[i];
  scale[i] = sc;
  shift[i] = be[i] - mu * sc;
}

// ---------------- WMMA GEMM kernels ----------------
// Block = 128 threads = 4 wave32, each wave computes a 16-row x 64-col tile.
// A-frag (f32 16x4): lanes 0-15 hold K=k+0,k+1 ; lanes 16-31 hold K=k+2,k+3.
// B-frag (f32 4x16): same striping over K. C/D: v8f, rows half*8+r, col = tile*16 + (lane&15).

// Z = A @ W + bias, with fused per-column sum / sum-of-squares (only rows < nValid)
__global__ void __launch_bounds__(128)
gemm_bias_stats(const float* __restrict__ A, const float* __restrict__ W,
                const float* __restrict__ bias, float* __restrict__ Z,
                float* __restrict__ gsum, float* __restrict__ gssq,
                int nPad, int nValid) {
  __shared__ float ssum[DH];
  __shared__ float sssq[DH];
  int tid = threadIdx.x;
  if (tid < DH) { ssum[tid] = 0.0f; sssq[tid] = 0.0f; }
  __syncthreads();

  int wave = tid >> 5, lane = tid & 31;
  int rowBase = blockIdx.x * 64 + wave * 16;
  if (rowBase < nPad) {
    int half = lane >> 4, l = lane & 15;
    const float* Ap = A + (size_t)(rowBase + l) * DH + half * 2;
    v8f c[4];
#pragma unroll
    for (int t = 0; t < 4; ++t) c[t] = (v8f){0,0,0,0,0,0,0,0};
#pragma unroll 4
    for (int k = 0; k < DH; k += 4) {
      v2f a;
      a.x = Ap[k];
      a.y = Ap[k + 1];
      const float* Wp = W + (size_t)(k + half * 2) * DH + l;
#pragma unroll
      for (int t = 0; t < 4; ++t) {
        v2f b;
        b.x = Wp[t * 16];
        b.y = Wp[DH + t * 16];
        c[t] = wmma_f32_k4(a, b, c[t]);
      }
    }
    int r0 = rowBase + half * 8;
#pragma unroll
    for (int t = 0; t < 4; ++t) {
      int col = t * 16 + l;
      float bv = bias[col];
      float s = 0.0f, q = 0.0f;
#pragma unroll
      for (int r = 0; r < 8; ++r) {
        float v = c[t][r] + bv;
        Z[(size_t)(r0 + r) * DH + col] = v;
        if (r0 + r < nValid) { s += v; q += v * v; }
      }
      atomicAdd(&ssum[col], s);
      atomicAdd(&sssq[col], q);
    }
  }
  __syncthreads();
  if (tid < DH) {
    atomicAdd(&gsum[tid], ssum[tid]);
    atomicAdd(&gssq[tid], sssq[tid]);
  }
}

// Out = LeakyReLU_0.1(Z*scale + shift) @ W + bias ; store rows < nStoreRows, cols < nStoreCols (padded W)
template <int NT, int LDW>
__global__ void __launch_bounds__(128)
bn_act_gemm(const float* __restrict__ Z, const float* __restrict__ scale,
            const float* __restrict__ shift, const float* __restrict__ W,
            const float* __restrict__ bias, float* __restrict__ Out,
            int nPad, int nStoreRows, int nStoreCols, int ldo) {
  int tid = threadIdx.x;
  int wave = tid >> 5, lane = tid & 31;
  int rowBase = blockIdx.x * 64 + wave * 16;
  if (rowBase >= nPad) return;
  int half = lane >> 4, l = lane & 15;
  const float* Zp = Z + (size_t)(rowBase + l) * DH + half * 2;
  v8f c[NT];
#pragma unroll
  for (int t = 0; t < NT; ++t) c[t] = (v8f){0,0,0,0,0,0,0,0};
#pragma unroll 4
  for (int k = 0; k < DH; k += 4) {
    int kc = k + half * 2;
    float z0 = Zp[k]     * scale[kc]     + shift[kc];
    float z1 = Zp[k + 1] * scale[kc + 1] + shift[kc + 1];
    v2f a;
    a.x = z0 > 0.0f ? z0 : 0.1f * z0;
    a.y = z1 > 0.0f ? z1 : 0.1f * z1;
    const float* Wp = W + (size_t)kc * LDW + l;
#pragma unroll
    for (int t = 0; t < NT; ++t) {
      v2f b;
      b.x = Wp[t * 16];
      b.y = Wp[LDW + t * 16];
      c[t] = wmma_f32_k4(a, b, c[t]);
    }
  }
  int r0 = rowBase + half * 8;
#pragma unroll
  for (int t = 0; t < NT; ++t) {
    int col = t * 16 + l;
    if (col >= nStoreCols) continue;
    float bv = bias[col];
#pragma unroll
    for (int r = 0; r < 8; ++r) {
      int row = r0 + r;
      if (row < nStoreRows) Out[(size_t)row * ldo + col] = c[t][r] + bv;
    }
  }
}

// ---------------- host side ----------------

struct MlpP {
  const float *W1, *b1, *g1, *be1, *W2, *b2;
};

extern "C" void kernel_launch(void* const* d_in, const int* in_sizes, int n_in,
                              void* d_out, int out_size, void* d_ws, size_t ws_size,
                              hipStream_t stream) {
  // --- locate inputs by size (pytree flatten order is ambiguous) ---
  int idx_x = -1, idx_edge = -1, idx_batch = -1;
  int pidx[36];
  int np = 0;
  for (int i = 0; i < n_in; ++i) {
    int s = in_sizes[i];
    if (s == N_NODES * DH)      idx_x = i;
    else if (s == 2 * N_EDGES)  idx_edge = i;
    else if (s == N_NODES)      idx_batch = i;
    else if ((s == DH * DH || s == DH) && np < 36) pidx[np++] = i;
  }
  const float* x     = (const float*)d_in[idx_x];
  const int*   edge  = (const int*)d_in[idx_edge];
  const int*   batch = (const int*)d_in[idx_batch];

  MlpP blk[6];
  for (int b = 0; b < 6; ++b) {
    const int* q = pidx + b * 6;
    bool sortedKeys = (in_sizes[q[1]] == DH * DH);  // [W1,W2,b1,b2,be1,g1] vs [W1,b1,g1,be1,W2,b2]
    if (sortedKeys) {
      blk[b].W1  = (const float*)d_in[q[0]];
      blk[b].W2  = (const float*)d_in[q[1]];
      blk[b].b1  = (const float*)d_in[q[2]];
      blk[b].b2  = (const float*)d_in[q[3]];
      blk[b].be1 = (const float*)d_in[q[4]];
      blk[b].g1  = (const float*)d_in[q[5]];
    } else {
      blk[b].W1  = (const float*)d_in[q[0]];
      blk[b].b1  = (const float*)d_in[q[1]];
      blk[b].g1  = (const float*)d_in[q[2]];
      blk[b].be1 = (const float*)d_in[q[3]];
      blk[b].W2  = (const float*)d_in[q[4]];
      blk[b].b2  = (const float*)d_in[q[5]];
    }
  }

  // --- workspace carve-up (floats) ---
  float* ws = (float*)d_ws;
  const size_t NODE_F = (size_t)N_NODES * DH;  // 6.4M
  float* hA     = ws;
  float* hB     = hA + NODE_F;
  float* agg    = hB + NODE_F;
  float* zb     = agg + NODE_F;
  float* pooled = zb + NODE_F;                 // 512*64
  float* w2pad  = pooled + (size_t)N_GRAPHS_PAD * DH;  // 64*16
  float* b2pad  = w2pad + DH * 16;             // 16
  float* gsum   = b2pad + 16;                  // 64
  float* gssq   = gsum + DH;                   // 64
  float* scl    = gssq + DH;                   // 64
  float* shf    = scl + DH;                    // 64

  const int nodeGrid = (N_NODES + 63) / 64;              // 1563 blocks of 4 waves
  const int copyGrid = (N_NODES * 16 + 255) / 256;       // float4 per 16 threads/node
  const int edgeGrid = (N_EDGES * 16 + 255) / 256;

  // --- GIN layers ---
  const float* hcur = x;
  for (int L = 0; L < 5; ++L) {
    float* hnext = (L & 1) ? hB : hA;
    const MlpP& p = blk[L];
    // agg = hcur (self term, eps=0), then agg[dst] += hcur[src]
    vec_copy4<<<copyGrid, 256, 0, stream>>>((float4*)agg, (const float4*)hcur, N_NODES * 16);
    edge_scatter<<<edgeGrid, 256, 0, stream>>>(edge, hcur, agg, N_EDGES);
    vec_zero<<<1, 128, 0, stream>>>(gsum, 2 * DH);
    gemm_bias_stats<<<nodeGrid, 128, 0, stream>>>(agg, p.W1, p.b1, zb, gsum, gssq,
                                                  N_NODES, N_NODES);
    bn_finalize<<<1, DH, 0, stream>>>(gsum, gssq, p.g1, p.be1, scl, shf, 1.0f / N_NODES);
    bn_act_gemm<4, DH><<<nodeGrid, 128, 0, stream>>>(zb, scl, shf, p.W2, p.b2, hnext,
                                                     N_NODES, N_NODES, DH, DH);
    hcur = hnext;
  }

  // --- global_add_pool (pad to 512 graphs with zeros) ---
  vec_zero<<<(N_GRAPHS_PAD * DH + 255) / 256, 256, 0, stream>>>(pooled, N_GRAPHS_PAD * DH);
  pool_scatter<<<copyGrid, 256, 0, stream>>>(batch, hcur, pooled, N_NODES);

  // --- classifier head MLP ---
  const MlpP& m = blk[5];
  pad_w2<<<4, 256, 0, stream>>>(m.W2, m.b2, w2pad, b2pad);
  vec_zero<<<1, 128, 0, stream>>>(gsum, 2 * DH);
  gemm_bias_stats<<<N_GRAPHS_PAD / 64, 128, 0, stream>>>(pooled, m.W1, m.b1, zb, gsum, gssq,
                                                         N_GRAPHS_PAD, N_GRAPHS);
  bn_finalize<<<1, DH, 0, stream>>>(gsum, gssq, m.g1, m.be1, scl, shf, 1.0f / N_GRAPHS);
  bn_act_gemm<1, 16><<<N_GRAPHS_PAD / 64, 128, 0, stream>>>(zb, scl, shf, w2pad, b2pad,
                                                            (float*)d_out, N_GRAPHS_PAD,
                                                            N_GRAPHS, 10, 10);
}